// EMAResidualVectorQuantizer_51608327028992
// MI455X (gfx1250) — compile-verified
//
#include <hip/hip_runtime.h>

typedef float v2f __attribute__((ext_vector_type(2)));
typedef float v8f __attribute__((ext_vector_type(8)));

#define BB 16
#define CC 128
#define TT 4096
#define QQ 8
#define KK 1024
#define NN (BB * TT)            // 65536 rows
#define ROWS_PER_BLOCK 64
#define THREADS 128             // 4 wave32
#define RSTRIDE 132             // padded LDS row stride (floats): 528B rows, 16B aligned, conflict-free
#define CHUNK 64                // codebook rows per async chunk (double buffered)
#define NCHUNK (KK / CHUNK)     // 16

#define QUANT_ELEMS (BB * CC * TT)   // 8388608
#define CODES_ELEMS (BB * QQ * TT)   // 524288
#define CODES_OFF QUANT_ELEMS
#define LOSS_OFF (QUANT_ELEMS + CODES_ELEMS)

#define EMB0_F (ROWS_PER_BLOCK * RSTRIDE)  // float offset of code buffers in LDS
#define BUF_F (CHUNK * RSTRIDE)            // floats per code buffer
#define SMEM_FLOATS (EMB0_F + 2 * BUF_F)   // ~101 KB -> 3 blocks per 320KB WGP

__global__ void rvq_init(float* out) {
  if (threadIdx.x == 0 && blockIdx.x == 0) {
    out[LOSS_OFF + 0] = 0.0f;  // commitment_loss accumulator
    out[LOSS_OFF + 1] = 0.0f;  // codebook_loss (always 0 in reference)
  }
}

// bias[q*K + k] = -0.5 * ||emb[q,k,:]||^2   (one wave32 per code row, coalesced float4)
__global__ __launch_bounds__(256) void rvq_bias(const float* __restrict__ emb,
                                                float* __restrict__ bias) {
  int gid  = blockIdx.x * blockDim.x + threadIdx.x;
  int code = gid >> 5;
  int lane = gid & 31;
  const float4* e4 = (const float4*)(emb + (size_t)code * CC);
  float4 v = e4[lane];
  float s = v.x * v.x + v.y * v.y + v.z * v.z + v.w * v.w;
#pragma unroll
  for (int m = 1; m < 32; m <<= 1) s += __shfl_xor(s, m, 32);
  if (lane == 0) bias[code] = -0.5f * s;
}

__global__ __launch_bounds__(THREADS) void rvq_main(const float* __restrict__ lat,
                                                    const float* __restrict__ emb,
                                                    const float* __restrict__ bias,
                                                    float* __restrict__ out) {
  extern __shared__ float sm[];
  float* res = sm;  // [64][RSTRIDE] residual tile

  const int tid  = threadIdx.x;
  const int n0   = blockIdx.x * ROWS_PER_BLOCK;  // 64 | T, so b is constant per block
  const int b    = n0 >> 12;
  const int t0   = n0 & (TT - 1);
  const int lane = tid & 31;
  const int half = lane >> 4;
  const int l15  = lane & 15;
  const int rowbase = (tid >> 5) * 16;  // 16 rows per wave

  // LDS base byte offset (addrspace(3) pointers carry the raw LDS offset in the low 32 bits)
  const unsigned smbase = (unsigned)(size_t)sm;

  // Issue one 64-code chunk as async global->LDS b128 transfers (tracked by ASYNCcnt).
  auto issue_chunk = [&](int qi, int kc, int bufsel) {
    unsigned lbase = smbase + (unsigned)((EMB0_F + bufsel * BUF_F) * 4);
    const float* gbase = emb + ((size_t)(qi * KK + kc * CHUNK)) * CC;
    for (int i = tid; i < CHUNK * (CC / 4); i += THREADS) {
      int code = i >> 5, c4 = i & 31;
      unsigned lo = lbase + (unsigned)((code * RSTRIDE + c4 * 4) * 4);
      unsigned long long ga = (unsigned long long)(size_t)(gbase + code * CC + c4 * 4);
      asm volatile("global_load_async_to_lds_b128 %0, %1, off"
                   :: "v"(lo), "v"(ga)
                   : "memory");
    }
  };

  // Kick off chunk (q=0, kc=0) into buffer 0, then overlap with the latents load.
  issue_chunk(0, 0, 0);

  // Load latents tile -> residual (coalesced along t)
  for (int i = tid; i < ROWS_PER_BLOCK * CC; i += THREADS) {
    int c = i >> 6, tl = i & 63;
    res[tl * RSTRIDE + c] = lat[((size_t)(b * CC + c)) * TT + t0 + tl];
  }

  float closs = 0.0f;

  for (int q = 0; q < QQ; ++q) {
    float bv_best[8];
    int   bi_best[8];
#pragma unroll
    for (int j = 0; j < 8; ++j) { bv_best[j] = -3.4e38f; bi_best[j] = 0; }

    // Chunk (q, 0) was issued in the previous stage (or prologue): drain + make visible.
    asm volatile("s_wait_asynccnt 0x0" ::: "memory");
    __syncthreads();

    for (int kc = 0; kc < NCHUNK; ++kc) {
      // Prefetch the next chunk while computing on the current one.
      if (kc + 1 < NCHUNK)      issue_chunk(q, kc + 1, (kc + 1) & 1);
      else if (q + 1 < QQ)      issue_chunk(q + 1, 0, 0);

      const float* ebuf = sm + EMB0_F + (kc & 1) * BUF_F;

#pragma unroll
      for (int kt = 0; kt < CHUNK / 16; ++kt) {
        const int col = kc * CHUNK + kt * 16 + l15;
        const float bb0 = bias[q * KK + col];  // -0.5*||e||^2 folded into accumulator
        v8f acc0, acc1;
#pragma unroll
        for (int j = 0; j < 8; ++j) { acc0[j] = bb0; acc1[j] = 0.0f; }

        // A: 16x4 residual tile (lane = row, half selects K pair); B: 4x16 code tile.
        // Two independent accumulator chains double WMMA ILP per wave.
        const float* arow = &res[(rowbase + l15) * RSTRIDE + half * 2];
        const float* brow = &ebuf[(kt * 16 + l15) * RSTRIDE + half * 2];
#pragma unroll 8
        for (int ck2 = 0; ck2 < CC / 8; ++ck2) {
          v2f a0 = *(const v2f*)(arow + ck2 * 8);
          v2f b0 = *(const v2f*)(brow + ck2 * 8);
          v2f a1 = *(const v2f*)(arow + ck2 * 8 + 4);
          v2f b1 = *(const v2f*)(brow + ck2 * 8 + 4);
          acc0 = __builtin_amdgcn_wmma_f32_16x16x4_f32(false, a0, false, b0,
                                                       (short)0, acc0, false, false);
          acc1 = __builtin_amdgcn_wmma_f32_16x16x4_f32(false, a1, false, b1,
                                                       (short)0, acc1, false, false);
        }
#pragma unroll
        for (int j = 0; j < 8; ++j) {
          float v = acc0[j] + acc1[j];
          if (v > bv_best[j]) { bv_best[j] = v; bi_best[j] = col; }
        }
      }

      if (kc + 1 < NCHUNK) {
        asm volatile("s_wait_asynccnt 0x0" ::: "memory");
        __syncthreads();
      }
    }

    // Argmax across the 16 column-lanes of each half (ties -> lowest index == argmin-first)
#pragma unroll
    for (int m = 1; m < 16; m <<= 1) {
#pragma unroll
      for (int j = 0; j < 8; ++j) {
        float ov = __shfl_xor(bv_best[j], m, 32);
        int   oi = __shfl_xor(bi_best[j], m, 32);
        if (ov > bv_best[j] || (ov == bv_best[j] && oi < bi_best[j])) {
          bv_best[j] = ov; bi_best[j] = oi;
        }
      }
    }

    // Codes: lane 0 holds rows 0..7, lane 16 holds rows 8..15
    if (l15 == 0) {
#pragma unroll
      for (int j = 0; j < 8; ++j) {
        int gr = rowbase + half * 8 + j;
        int n = n0 + gr, bb = n >> 12, tt = n & (TT - 1);
        out[CODES_OFF + ((size_t)(bb * QQ + q)) * TT + tt] = (float)bi_best[j];
      }
    }

    // Gather winning rows (coalesced), update residual, accumulate ||r_new||^2 (= stage loss).
    // This phase overlaps the in-flight async load of chunk (q+1, 0).
    float sumsq = 0.0f;
    const float4* e4 = (const float4*)emb;
#pragma unroll
    for (int r = 0; r < 16; ++r) {
      int idxr = (r < 8) ? __shfl(bi_best[r], 0, 32) : __shfl(bi_best[r - 8], 16, 32);
      float4 ev = e4[((size_t)(q * KK + idxr)) * (CC / 4) + lane];
      float4* rp = (float4*)&res[(rowbase + r) * RSTRIDE + lane * 4];
      float4 rv = *rp;
      rv.x -= ev.x; rv.y -= ev.y; rv.z -= ev.z; rv.w -= ev.w;
      *rp = rv;
      sumsq += rv.x * rv.x + rv.y * rv.y + rv.z * rv.z + rv.w * rv.w;
    }
    closs += sumsq;
  }

  // Wave-reduce loss, one scaled f32 atomic per wave
#pragma unroll
  for (int m = 1; m < 32; m <<= 1) closs += __shfl_xor(closs, m, 32);
  if (lane == 0)
    atomicAdd(&out[LOSS_OFF],
              closs * (1.0f / ((float)NN * (float)CC * (float)QQ)));

  __syncthreads();
  // quantized = latents - final residual (coalesced along t)
  for (int i = tid; i < ROWS_PER_BLOCK * CC; i += THREADS) {
    int c = i >> 6, tl = i & 63;
    size_t gi = ((size_t)(b * CC + c)) * TT + t0 + tl;
    out[gi] = lat[gi] - res[tl * RSTRIDE + c];
  }
}

extern "C" void kernel_launch(void* const* d_in, const int* in_sizes, int n_in,
                              void* d_out, int out_size, void* d_ws, size_t ws_size,
                              hipStream_t stream) {
  const float* lat = (const float*)d_in[0];   // [B, C, T]
  const float* emb = (const float*)d_in[1];   // [Q, K, C]
  float* out  = (float*)d_out;
  float* bias = (float*)d_ws;                 // Q*K floats = 32 KB

  rvq_init<<<1, 1, 0, stream>>>(out);
  rvq_bias<<<(QQ * KK * 32) / 256, 256, 0, stream>>>(emb, bias);
  rvq_main<<<NN / ROWS_PER_BLOCK, THREADS, SMEM_FLOATS * sizeof(float), stream>>>(
      lat, emb, bias, out);
}